// MyGCN2_24180665876563
// MI455X (gfx1250) — compile-verified
//
#include <hip/hip_runtime.h>
#include <hip/hip_bf16.h>

typedef __attribute__((ext_vector_type(2))) float v2f;
typedef __attribute__((ext_vector_type(8))) float v8f;

#define F_DIM 128
#define C_DIM 40

// ---------------- degree / normalization ----------------
__global__ void k_deg_init(float* __restrict__ deg, int N) {
  int i = blockIdx.x * blockDim.x + threadIdx.x;
  if (i < N) deg[i] = 1.0f;  // self loop
}

__global__ void k_deg_edges(const int* __restrict__ ei, float* __restrict__ deg, int E) {
  int e = blockIdx.x * blockDim.x + threadIdx.x;
  if (e < E) atomicAdd(&deg[ei[E + e]], 1.0f);  // dst = edge_index[1][e]
}

__global__ void k_dinv(float* __restrict__ deg, int N) {
  int i = blockIdx.x * blockDim.x + threadIdx.x;
  if (i < N) deg[i] = rsqrtf(deg[i]);  // deg >= 1 always
}

// ---------------- GEMM1: hx = x @ W1   (N x 128) @ (128 x 128) ----------------
// blockDim = 256 (8 waves). Each wave: 16x16 output tile; block: 16 rows x 128 cols.
__global__ void k_gemm_xW1(const float* __restrict__ X, const float* __restrict__ W,
                           float* __restrict__ Hx, int N) {
  const int wave = threadIdx.x >> 5;
  const int lane = threadIdx.x & 31;
  const int m    = lane & 15;   // A row / B-C-D col within tile
  const int kh   = lane >> 4;   // K half select
  const int n0   = wave * 16;
  const int row0 = blockIdx.x * 16;

  v8f acc = {};
  const int rowA = min(row0 + m, N - 1);
  const float* xrow = X + (size_t)rowA * F_DIM;
  for (int k0 = 0; k0 < F_DIM; k0 += 4) {
    v2f a, b;
    a.x = xrow[k0 + 2 * kh];
    a.y = xrow[k0 + 2 * kh + 1];
    b.x = W[(size_t)(k0 + 2 * kh) * F_DIM + n0 + m];
    b.y = W[(size_t)(k0 + 2 * kh + 1) * F_DIM + n0 + m];
    acc = __builtin_amdgcn_wmma_f32_16x16x4_f32(false, a, false, b, (short)0, acc,
                                                false, false);
  }
  const int n = n0 + m;
  float* outp = Hx + (size_t)(row0 + 8 * kh) * F_DIM + n;
  if (row0 + 16 <= N) {  // full tile: branch-free epilogue (all blocks when 16|N)
#pragma unroll
    for (int v = 0; v < 8; ++v) outp[(size_t)v * F_DIM] = acc[v];
  } else {
#pragma unroll
    for (int v = 0; v < 8; ++v)
      if (row0 + 8 * kh + v < N) outp[(size_t)v * F_DIM] = acc[v];
  }
}

// ---------------- self-loop + bias init: agg = b + h*dinv^2 ----------------
__global__ void k_self_bias_128(const float* __restrict__ Hx, const float* __restrict__ b,
                                const float* __restrict__ dinv, float* __restrict__ Agg,
                                int NF) {
  int idx = blockIdx.x * blockDim.x + threadIdx.x;
  if (idx >= NF) return;
  int i = idx >> 7;   // /128
  int f = idx & 127;
  float di = dinv[i];
  Agg[idx] = b[f] + Hx[idx] * di * di;
}

__global__ void k_self_bias_40(const float* __restrict__ Hy, const float* __restrict__ b,
                               const float* __restrict__ dinv, float* __restrict__ Agg,
                               int NC) {
  int idx = blockIdx.x * blockDim.x + threadIdx.x;
  if (idx >= NC) return;
  int i = idx / C_DIM;
  int c = idx - i * C_DIM;
  float di = dinv[i];
  Agg[idx] = b[c] + Hy[idx] * di * di;
}

// ---------------- edge scatter, 128 features: one wave32 per edge ----------------
__global__ void k_scatter128(const int* __restrict__ ei, const float* __restrict__ Hx,
                             const float* __restrict__ dinv, float* __restrict__ Agg,
                             int E) {
  int wid  = (blockIdx.x * blockDim.x + threadIdx.x) >> 5;
  int lane = threadIdx.x & 31;
  if (wid >= E) return;
  int s = ei[wid];
  int d = ei[E + wid];
  float nrm = dinv[s] * dinv[d];
  const float4* srow = (const float4*)(Hx + (size_t)s * F_DIM);
  float4 v = srow[lane];                       // 16B/lane -> full 512B row per wave
  float* drow = Agg + (size_t)d * F_DIM + lane * 4;
  atomicAdd(drow + 0, v.x * nrm);
  atomicAdd(drow + 1, v.y * nrm);
  atomicAdd(drow + 2, v.z * nrm);
  atomicAdd(drow + 3, v.w * nrm);
}

// ---------------- edge scatter, 40 features: one wave32 per edge ----------------
__global__ void k_scatter40(const int* __restrict__ ei, const float* __restrict__ Hy,
                            const float* __restrict__ dinv, float* __restrict__ Agg,
                            int E) {
  int wid  = (blockIdx.x * blockDim.x + threadIdx.x) >> 5;
  int lane = threadIdx.x & 31;
  if (wid >= E) return;
  int s = ei[wid];
  int d = ei[E + wid];
  float nrm = dinv[s] * dinv[d];
  const float* srow = Hy + (size_t)s * C_DIM;
  float* drow = Agg + (size_t)d * C_DIM;
  atomicAdd(drow + lane, srow[lane] * nrm);
  if (lane < C_DIM - 32) atomicAdd(drow + 32 + lane, srow[32 + lane] * nrm);
}

// ---------------- GEMM2: hy = relu(agg1) @ W2   (N x 128) @ (128 x 40) ----------------
// blockDim = 96 (3 waves): col tiles 0,16,32 (cols >= 40 guarded/zero-padded)
__global__ void k_gemm_h1W2(const float* __restrict__ Agg1, const float* __restrict__ W2,
                            float* __restrict__ Hy, int N) {
  const int wave = threadIdx.x >> 5;
  const int lane = threadIdx.x & 31;
  const int m    = lane & 15;
  const int kh   = lane >> 4;
  const int n0   = wave * 16;
  const int row0 = blockIdx.x * 16;
  const int n    = n0 + m;
  const bool nok = (n < C_DIM);

  v8f acc = {};
  const int rowA = min(row0 + m, N - 1);
  const float* arow = Agg1 + (size_t)rowA * F_DIM;
  for (int k0 = 0; k0 < F_DIM; k0 += 4) {
    v2f a, b;
    a.x = fmaxf(arow[k0 + 2 * kh], 0.0f);       // fused ReLU on A
    a.y = fmaxf(arow[k0 + 2 * kh + 1], 0.0f);
    b.x = nok ? W2[(size_t)(k0 + 2 * kh) * C_DIM + n] : 0.0f;
    b.y = nok ? W2[(size_t)(k0 + 2 * kh + 1) * C_DIM + n] : 0.0f;
    acc = __builtin_amdgcn_wmma_f32_16x16x4_f32(false, a, false, b, (short)0, acc,
                                                false, false);
  }
  if (nok) {
    float* outp = Hy + (size_t)(row0 + 8 * kh) * C_DIM + n;
    if (row0 + 16 <= N) {
#pragma unroll
      for (int v = 0; v < 8; ++v) outp[(size_t)v * C_DIM] = acc[v];
    } else {
#pragma unroll
      for (int v = 0; v < 8; ++v)
        if (row0 + 8 * kh + v < N) outp[(size_t)v * C_DIM] = acc[v];
    }
  }
}

// ---------------- output h2 = relu(agg2) ----------------
__global__ void k_relu_out(const float* __restrict__ Agg2, float* __restrict__ Out,
                           int total) {
  int idx = blockIdx.x * blockDim.x + threadIdx.x;
  if (idx < total) Out[idx] = fmaxf(Agg2[idx], 0.0f);
}

// ---------------- GEMM3: out = relu(agg2) @ Wlin + blin   (N x 40) @ (40 x 40) ----------------
__global__ void k_gemm_h2Wlin(const float* __restrict__ Agg2, const float* __restrict__ Wl,
                              const float* __restrict__ bl, float* __restrict__ Out,
                              int N) {
  const int wave = threadIdx.x >> 5;
  const int lane = threadIdx.x & 31;
  const int m    = lane & 15;
  const int kh   = lane >> 4;
  const int n0   = wave * 16;
  const int row0 = blockIdx.x * 16;
  const int n    = n0 + m;
  const bool nok = (n < C_DIM);

  v8f acc = {};
  const int rowA = min(row0 + m, N - 1);
  const float* arow = Agg2 + (size_t)rowA * C_DIM;
  for (int k0 = 0; k0 < C_DIM; k0 += 4) {      // K = 40, exactly 10 steps
    v2f a, b;
    a.x = fmaxf(arow[k0 + 2 * kh], 0.0f);      // fused ReLU on A
    a.y = fmaxf(arow[k0 + 2 * kh + 1], 0.0f);
    b.x = nok ? Wl[(size_t)(k0 + 2 * kh) * C_DIM + n] : 0.0f;
    b.y = nok ? Wl[(size_t)(k0 + 2 * kh + 1) * C_DIM + n] : 0.0f;
    acc = __builtin_amdgcn_wmma_f32_16x16x4_f32(false, a, false, b, (short)0, acc,
                                                false, false);
  }
  if (nok) {
    float bias = bl[n];
    float* outp = Out + (size_t)(row0 + 8 * kh) * C_DIM + n;
    if (row0 + 16 <= N) {
#pragma unroll
      for (int v = 0; v < 8; ++v) outp[(size_t)v * C_DIM] = acc[v] + bias;
    } else {
#pragma unroll
      for (int v = 0; v < 8; ++v)
        if (row0 + 8 * kh + v < N) outp[(size_t)v * C_DIM] = acc[v] + bias;
    }
  }
}

// ---------------- host side ----------------
static inline size_t align256(size_t x) { return (x + 255) & ~(size_t)255; }

extern "C" void kernel_launch(void* const* d_in, const int* in_sizes, int n_in,
                              void* d_out, int out_size, void* d_ws, size_t ws_size,
                              hipStream_t stream) {
  const float* x    = (const float*)d_in[0];
  const int*   ei   = (const int*)d_in[1];
  const float* W1   = (const float*)d_in[2];
  const float* b1   = (const float*)d_in[3];
  const float* W2   = (const float*)d_in[4];
  const float* b2   = (const float*)d_in[5];
  const float* Wlin = (const float*)d_in[6];
  const float* blin = (const float*)d_in[7];
  float* out = (float*)d_out;

  const int N = in_sizes[0] / F_DIM;
  const int E = in_sizes[1] / 2;

  // workspace layout
  char* ws = (char*)d_ws;
  float* dinv = (float*)ws;  ws += align256((size_t)N * sizeof(float));
  float* hx   = (float*)ws;  ws += align256((size_t)N * F_DIM * sizeof(float));
  float* agg1 = (float*)ws;  ws += align256((size_t)N * F_DIM * sizeof(float));
  float* hy   = (float*)ws;  ws += align256((size_t)N * C_DIM * sizeof(float));
  float* agg2 = (float*)ws;  ws += align256((size_t)N * C_DIM * sizeof(float));

  const int NF = N * F_DIM;   // 6.4M, fits int
  const int NC = N * C_DIM;   // 2.0M, fits int

  // 1) normalization coefficients
  k_deg_init<<<(N + 255) / 256, 256, 0, stream>>>(dinv, N);
  k_deg_edges<<<(E + 255) / 256, 256, 0, stream>>>(ei, dinv, E);
  k_dinv<<<(N + 255) / 256, 256, 0, stream>>>(dinv, N);

  // 2) hx = x @ W1 (WMMA f32)
  k_gemm_xW1<<<(N + 15) / 16, 256, 0, stream>>>(x, W1, hx, N);

  // 3) layer-1 aggregation
  k_self_bias_128<<<(NF + 255) / 256, 256, 0, stream>>>(hx, b1, dinv, agg1, NF);
  k_scatter128<<<(E + 7) / 8, 256, 0, stream>>>(ei, hx, dinv, agg1, E);

  // 4) hy = relu(agg1) @ W2 (WMMA f32, fused ReLU)
  k_gemm_h1W2<<<(N + 15) / 16, 96, 0, stream>>>(agg1, W2, hy, N);

  // 5) layer-2 aggregation
  k_self_bias_40<<<(NC + 255) / 256, 256, 0, stream>>>(hy, b2, dinv, agg2, NC);
  k_scatter40<<<(E + 7) / 8, 256, 0, stream>>>(ei, hy, dinv, agg2, E);

  // 6) outputs: h2 = relu(agg2), out = h2 @ Wlin + blin
  k_relu_out<<<(NC + 255) / 256, 256, 0, stream>>>(agg2, out, NC);
  k_gemm_h2Wlin<<<(N + 15) / 16, 96, 0, stream>>>(agg2, Wlin, blin, out + NC, N);
}